// FasterRCNN_24970939859182
// MI455X (gfx1250) — compile-verified
//
#include <hip/hip_runtime.h>
#include <cstdint>
#include <cstddef>

#define NA      90000
#define NGT     256
#define KNMS    2000
#define CAP     2048
#define NMS_THR 0.7f
#define IMG     1600.0f

typedef unsigned int v4u __attribute__((ext_vector_type(4)));
typedef int          v4i __attribute__((ext_vector_type(4)));
typedef int          v8i __attribute__((ext_vector_type(8)));

// ---------------------------------------------------------------------------
// Closed-form anchor for flat index a (replicates _anchor_mesh exactly,
// including the quirky voc->yolo axis mixing), clipped to [0, IMG].
// ---------------------------------------------------------------------------
__device__ __forceinline__ void anchor_box(int a, float& x1, float& y1,
                                           float& x2, float& y2)
{
    const int k    = a % 9;
    const int cell = a / 9;
    const float sx = (float)(cell % 100) * 16.0f;
    const float sy = (float)(cell / 100) * 16.0f;

    const int ri = k / 3, si = k % 3;
    const float r  = (ri == 0) ? 0.5f : ((ri == 1) ? 1.0f : 2.0f);
    const float sc = (si == 0) ? 8.0f : ((si == 1) ? 16.0f : 32.0f);

    const float S   = 16.0f;              // stride = 1600/100
    const float ycx = 0.5f * (S - 2.0f);  // base yolo cx = cy = 7
    const float wr  = rintf(sqrtf(S * S / r));
    const float hr  = rintf(wr * r);
    const float rx1 = ycx - 0.5f * (wr - 1.0f);
    const float ry1 = ycx - 0.5f * (hr - 1.0f);
    const float rx2 = ycx + 0.5f * (wr - 1.0f);
    const float ry2 = ycx + 0.5f * (hr - 1.0f);
    // quirky voc->yolo (x/y index mix), replicated faithfully:
    const float acx = rx1 + 0.5f * (ry2 - 1.0f);
    const float acy = ry1 + 0.5f * (rx2 - 1.0f);
    const float aw  = ry2 - ry1 + 1.0f;
    const float ah  = rx2 - rx1 + 1.0f;
    const float w2  = 0.5f * (aw * sc - 1.0f);
    const float h2  = 0.5f * (ah * sc - 1.0f);
    x1 = fminf(fmaxf(acx - w2 + sx, 0.0f), IMG);
    y1 = fminf(fmaxf(acy - h2 + sy, 0.0f), IMG);
    x2 = fminf(fmaxf(acx + w2 + sx, 0.0f), IMG);
    y2 = fminf(fmaxf(acy + h2 + sy, 0.0f), IMG);
}

// ---------------------------------------------------------------------------
// K1: IoU target assignment + proposal decode. GT table (4 KB) staged into
// LDS via the CDNA5 Tensor Data Mover (TENSORcnt path); float4 layout so the
// 256-iteration reduction reads with ds_load_b128 broadcasts.
// ---------------------------------------------------------------------------
__global__ __launch_bounds__(256) void k_targets_decode(
    const float* __restrict__ gt,      // [NGT,4]
    const float* __restrict__ deltas,  // [NA,4]
    float* __restrict__ out)           // [NA,9]
{
    __shared__ float4 sgt[NGT];

#if defined(__HIP_DEVICE_COMPILE__) &&                                        \
    __has_builtin(__builtin_amdgcn_tensor_load_to_lds) &&                     \
    __has_builtin(__builtin_amdgcn_s_wait_tensorcnt)
    if (threadIdx.x < 32u) {  // one wave issues the TDM descriptor
        const unsigned long long ga  = (unsigned long long)(uintptr_t)gt;
        const unsigned int       lds = (unsigned int)(uintptr_t)(void*)sgt;
        const unsigned int       d0  = NGT * 4;  // 1024 fp32 elements
        v4u g0 = { 1u,                                           // count=1
                   lds,                                          // lds_addr
                   (unsigned int)(ga & 0xFFFFFFFFull),           // gaddr lo
                   (unsigned int)((ga >> 32) & 0x01FFFFFFull) |
                       (2u << 30) };                             // gaddr hi | type=2
        v8i g1 = { (int)(2u << 16),                              // data_size=4B
                   (int)((d0 & 0xFFFFu) << 16),                  // tensor_dim0 lo16
                   (int)(((d0 >> 16) & 0xFFFFu) | (1u << 16)),   // dim0 hi | tensor_dim1=1
                   (int)((d0 & 0xFFFFu) << 16),                  // tile_dim0=1024
                   (int)1,                                       // tile_dim1=1
                   (int)d0,                                      // tensor_dim0_stride
                   0, 0 };
        v4i z4 = { 0, 0, 0, 0 };
#if __clang_major__ >= 23
        v8i z8 = { 0, 0, 0, 0, 0, 0, 0, 0 };
        __builtin_amdgcn_tensor_load_to_lds(g0, g1, z4, z4, z8, 0);
#else
        __builtin_amdgcn_tensor_load_to_lds(g0, g1, z4, z4, 0);
#endif
        __builtin_amdgcn_s_wait_tensorcnt(0);
    }
#else
    for (int i = threadIdx.x; i < NGT; i += blockDim.x)
        sgt[i] = ((const float4*)gt)[i];
#endif
    __syncthreads();

    const int a = blockIdx.x * blockDim.x + threadIdx.x;
    if (a >= NA) return;

    __builtin_prefetch(deltas + (size_t)a * 4, 0, 0);  // global_prefetch_b8

    float ax1, ay1, ax2, ay2;
    anchor_box(a, ax1, ay1, ax2, ay2);
    const float s_w = ax2 - ax1, s_h = ay2 - ay1;
    const float areaA = s_w * s_h;

    // per-anchor max/argmax IoU over 256 GT boxes (ds_load_b128 broadcasts)
    float best = -1.0f; int bi = 0;
    for (int g = 0; g < NGT; ++g) {
        const float4 gb = sgt[g];
        const float areaB = (gb.z - gb.x) * (gb.w - gb.y);
        const float iw = fmaxf(fminf(ax2, gb.z) - fmaxf(ax1, gb.x), 0.0f);
        const float ih = fmaxf(fminf(ay2, gb.w) - fmaxf(ay1, gb.y), 0.0f);
        const float inter = iw * ih;
        const float iou = inter / (areaA + areaB - inter);
        if (iou > best) { best = iou; bi = g; }
    }

    // parameterize(anchor, matched_gt)
    const float4 gb = sgt[bi];
    const float s_cx = 0.5f * (ax1 + ax2), s_cy = 0.5f * (ay1 + ay2);
    const float d_cx = 0.5f * (gb.x + gb.z), d_cy = 0.5f * (gb.y + gb.w);
    const float d_w = gb.z - gb.x, d_h = gb.w - gb.y;

    // decode + clip proposals (global_load_b128)
    const float4 dl = ((const float4*)deltas)[a];
    const float pcx = dl.x * s_w + s_cx, pcy = dl.y * s_h + s_cy;
    const float pw = expf(dl.z) * s_w,  ph = expf(dl.w) * s_h;

    float* row = out + (size_t)a * 9;
    row[0] = (s_cx - d_cx) / d_w;
    row[1] = (s_cy - d_cy) / d_h;
    row[2] = logf(s_w / d_w);
    row[3] = logf(s_h / d_h);
    row[4] = fminf(fmaxf(pcx - 0.5f * pw, 0.0f), IMG);
    row[5] = fminf(fmaxf(pcy - 0.5f * ph, 0.0f), IMG);
    row[6] = fminf(fmaxf(pcx + 0.5f * pw, 0.0f), IMG);
    row[7] = fminf(fmaxf(pcy + 0.5f * ph, 0.0f), IMG);
    row[8] = best;
}

// ---------------------------------------------------------------------------
// K2 (fused, single block): exact top-2000 radix select -> LDS compaction ->
// LDS bitonic sort (descending, exact lax.top_k tie order via ~idx key) ->
// candidate gather -> greedy NMS. All intermediate state lives in LDS
// (~65 KB of the 320 KB WGP LDS); zero global scratch, zero extra launches.
// ---------------------------------------------------------------------------
__global__ __launch_bounds__(1024) void k_topk_nms(
    const float* __restrict__ scores,    // [NA]
    const float* __restrict__ combined,  // [NA,9], boxes in cols 4..7
    float* __restrict__ keep_out)        // [KNMS]
{
    __shared__ unsigned int hist[256];
    __shared__ unsigned int sh_prefix, sh_remaining;
    __shared__ int sh_count;
    __shared__ unsigned long long sk[CAP];
    __shared__ float X1[KNMS], Y1[KNMS], X2[KNMS], Y2[KNMS], AR[KNMS];
    __shared__ int   KP[KNMS];
    const int t = threadIdx.x;

    // ---- init -------------------------------------------------------------
    if (t == 0) { sh_prefix = 0u; sh_remaining = KNMS; sh_count = 0; }
    for (int i = t; i < CAP; i += 1024) sk[i] = 0ull;
    __syncthreads();

    // ---- 4-pass MSD radix select: exact 2000th-largest score bits ---------
    const unsigned int hmask_tab[4] = { 0u, 0xFF000000u, 0xFFFF0000u, 0xFFFFFF00u };
    for (int pass = 0; pass < 4; ++pass) {
        const int shift = 24 - 8 * pass;
        const unsigned int hmask = hmask_tab[pass];
        if (t < 256) hist[t] = 0u;
        __syncthreads();
        const unsigned int prefix = sh_prefix;
        for (int i = t; i < NA; i += 1024) {
            const unsigned int b = __float_as_uint(scores[i]);
            if (((b ^ prefix) & hmask) == 0u)
                atomicAdd(&hist[(b >> shift) & 0xFFu], 1u);
        }
        __syncthreads();
        if (t == 0) {
            unsigned int cum = 0; int bsel = 0;
            for (int b = 255; b >= 0; --b) {
                const unsigned int c = hist[b];
                if (cum + c >= sh_remaining) { bsel = b; break; }
                cum += c;
            }
            sh_remaining -= cum;
            sh_prefix |= ((unsigned int)bsel) << shift;
        }
        __syncthreads();
    }
    const unsigned int T = sh_prefix;

    // ---- compact (scoreBits<<32 | ~idx) keys >= threshold into LDS --------
    for (int i = t; i < NA; i += 1024) {
        const unsigned int b = __float_as_uint(scores[i]);
        if (b >= T) {
            const int pos = atomicAdd(&sh_count, 1);
            if (pos < CAP)
                sk[pos] = ((unsigned long long)b << 32) |
                          (unsigned long long)(0xFFFFFFFFu - (unsigned int)i);
        }
    }
    __syncthreads();

    // ---- bitonic sort, descending (exact top_k order, ties by lower idx) --
    for (int k = 2; k <= CAP; k <<= 1) {
        for (int j = k >> 1; j > 0; j >>= 1) {
            for (int i = t; i < CAP; i += 1024) {
                const int p = i ^ j;
                if (p > i) {
                    const unsigned long long x = sk[i], y = sk[p];
                    const bool desc = ((i & k) == 0);
                    if (desc ? (x < y) : (x > y)) { sk[i] = y; sk[p] = x; }
                }
            }
            __syncthreads();
        }
    }

    // ---- gather top-2000 candidate boxes into LDS -------------------------
    for (int i = t; i < KNMS; i += 1024) {
        unsigned int idx = 0xFFFFFFFFu - (unsigned int)(sk[i] & 0xFFFFFFFFull);
        if (idx >= NA) idx = 0u;  // padding guard
        const float* r = combined + (size_t)idx * 9 + 4;
        const float x1 = r[0], y1 = r[1], x2 = r[2], y2 = r[3];
        X1[i] = x1; Y1[i] = y1; X2[i] = x2; Y2[i] = y2;
        AR[i] = (x2 - x1) * (y2 - y1);
        KP[i] = 1;
    }
    __syncthreads();

    // ---- greedy NMS (reference fori_loop semantics) -----------------------
    for (int i = 0; i < KNMS; ++i) {
        if (KP[i]) {
            const float x1 = X1[i], y1 = Y1[i], x2 = X2[i], y2 = Y2[i], ai = AR[i];
            for (int j = i + 1 + t; j < KNMS; j += 1024) {
                const float iw = fmaxf(0.0f, fminf(x2, X2[j]) - fmaxf(x1, X1[j]));
                const float ih = fmaxf(0.0f, fminf(y2, Y2[j]) - fmaxf(y1, Y1[j]));
                const float inter = iw * ih;
                const float iou = inter / (ai + AR[j] - inter);
                if (iou > NMS_THR) KP[j] = 0;
            }
        }
        __syncthreads();
    }
    for (int i = t; i < KNMS; i += 1024) keep_out[i] = KP[i] ? 1.0f : 0.0f;
}

// ---------------------------------------------------------------------------
extern "C" void kernel_launch(void* const* d_in, const int* in_sizes, int n_in,
                              void* d_out, int out_size, void* d_ws, size_t ws_size,
                              hipStream_t stream)
{
    (void)in_sizes; (void)n_in; (void)out_size; (void)d_ws; (void)ws_size;
    // inputs: 0=images (shape only), 1=feature_maps (shape only),
    //         2=true_bx [256,4], 3=rpn_deltas [90000,4], 4=rpn_scores [90000]
    const float* gt     = (const float*)d_in[2];
    const float* deltas = (const float*)d_in[3];
    const float* scores = (const float*)d_in[4];
    float* out = (float*)d_out;  // [90000*9] combined, then [2000] keep

    k_targets_decode<<<(NA + 255) / 256, 256, 0, stream>>>(gt, deltas, out);
    k_topk_nms      <<<1, 1024, 0, stream>>>(scores, out, out + (size_t)NA * 9);
}